// RetinaNetLoss_23845658428391
// MI455X (gfx1250) — compile-verified
//
#include <hip/hip_runtime.h>
#include <hip/hip_bf16.h>

#define NCLS 80
#define NB 8
#define NA 98304
#define NG 32

#define F_ALPHA 0.25f
#define F_BETA 0.11f
#define POS_THR 0.5f
#define NEG_THR 0.4f

typedef __attribute__((ext_vector_type(2))) float v2f;
typedef __attribute__((ext_vector_type(4))) float v4f;
typedef __attribute__((ext_vector_type(8))) float v8f;

// Full-wave (32-lane) sum using the matrix unit.
// A (16x4 f32, 2 VGPRs): v0 = per-lane value, v1 = 0.
// B (4x16 f32) = ones.  D[m][n] = A[m][0]+A[m][2] = val[m] + val[m+16].
// D layout: c[r] holds row r (lanes 0-15) / row r+8 (lanes 16-31), so
// sum(c[0..7]) gives lanes<16 = sum(val[0..7]+val[16..23]) and lanes>=16 the
// complementary half; one xor-16 shuffle completes the reduction.
__device__ __forceinline__ float wave_sum_wmma(float v) {
    v2f a;
    a[0] = v;
    a[1] = 0.0f;
    v2f ones;
    ones[0] = 1.0f;
    ones[1] = 1.0f;
    v8f c = {};
    c = __builtin_amdgcn_wmma_f32_16x16x4_f32(false, a, false, ones,
                                              (short)0, c, false, false);
    float s = ((c[0] + c[1]) + (c[2] + c[3])) + ((c[4] + c[5]) + (c[6] + c[7]));
    s += __shfl_xor(s, 16, 32);
    return s;
}

__global__ void retina_init_kernel(float* accum) {
    if (threadIdx.x < 4) accum[threadIdx.x] = 0.0f;
}

__global__ __launch_bounds__(256) void retina_loss_kernel(
        const float* __restrict__ cls_pred,   // [B, A, 80]
        const float* __restrict__ reg_pred,   // [B, A, 4]
        const float* __restrict__ anchors,    // [A, 4] xyxy
        const float* __restrict__ gt_boxes,   // [B, G, 4]
        const int*   __restrict__ gt_labels,  // [B, G] (1..80)
        float* __restrict__ accum) {          // [cls_sum, reg_sum, num_pos]
    __shared__ float4 s_gt[NG];
    __shared__ float  s_area[NG];
    __shared__ int    s_lab[NG];
    __shared__ float  s_wcls[8], s_wreg[8], s_wnp[8];

    const int tid = threadIdx.x;
    const int blocks_per_b = NA / 256;                 // 384
    const int b = blockIdx.x / blocks_per_b;
    const int a = (blockIdx.x % blocks_per_b) * 256 + tid;

    if (tid < NG) {
        float4 g = ((const float4*)gt_boxes)[b * NG + tid];
        s_gt[tid] = g;
        s_area[tid] = (g.z - g.x) * (g.w - g.y);
        s_lab[tid] = gt_labels[b * NG + tid];
    }
    __syncthreads();

    // ---- matching: max/argmax IoU over 32 gt boxes ----
    const float4 an = ((const float4*)anchors)[a];
    const float aw = an.z - an.x;
    const float ah = an.w - an.y;
    const float area1 = aw * ah;

    float best = -1.0f;
    int bi = 0;
#pragma unroll 4
    for (int g = 0; g < NG; ++g) {
        float4 gb = s_gt[g];
        float iw = fminf(an.z, gb.z) - fmaxf(an.x, gb.x);
        float ih = fminf(an.w, gb.w) - fmaxf(an.y, gb.y);
        iw = fmaxf(iw, 0.0f);
        ih = fmaxf(ih, 0.0f);
        float inter = iw * ih;
        float iou = inter * __builtin_amdgcn_rcpf(area1 + s_area[g] - inter);
        if (iou > best) { best = iou; bi = g; }  // first-max like argmax
    }
    const bool pos = best >= POS_THR;
    const bool valid = pos || (best < NEG_THR);
    const int tgt = pos ? (s_lab[bi] - 1) : -1;  // class index, -1 = background

    // ---- regression: encode matched box, smooth-L1 (pos anchors only) ----
    float reg_part = 0.0f;
    if (pos) {
        float4 m = s_gt[bi];
        float inv_aw = __builtin_amdgcn_rcpf(aw);
        float inv_ah = __builtin_amdgcn_rcpf(ah);
        float ax = an.x + 0.5f * aw, ay = an.y + 0.5f * ah;
        float gw = m.z - m.x, gh = m.w - m.y;
        float gx = m.x + 0.5f * gw, gy = m.y + 0.5f * gh;
        float t[4];
        t[0] = (gx - ax) * inv_aw;
        t[1] = (gy - ay) * inv_ah;
        t[2] = __logf(gw * inv_aw);
        t[3] = __logf(gh * inv_ah);
        v4f rp = __builtin_nontemporal_load(
            (const v4f*)(reg_pred + ((size_t)b * NA + a) * 4));
#pragma unroll
        for (int i = 0; i < 4; ++i) {
            float d = fabsf(rp[i] - t[i]);
            reg_part += (d < F_BETA) ? (0.5f / F_BETA) * d * d
                                     : d - 0.5f * F_BETA;
        }
    }

    // ---- classification: sigmoid focal loss over 80 classes ----
    float cls_part = 0.0f;
    if (valid) {
        const float* xp = cls_pred + ((size_t)b * NA + a) * NCLS;
#pragma unroll 5
        for (int c0 = 0; c0 < NCLS; c0 += 4) {
            v4f x4 = __builtin_nontemporal_load((const v4f*)(xp + c0));
#pragma unroll
            for (int j = 0; j < 4; ++j) {
                float x = x4[j];
                float e = __expf(-fabsf(x));           // exp(-|x|)
                float l1p = __logf(1.0f + e);          // log1p(exp(-|x|))
                float sp_pos = fmaxf(x, 0.0f) + l1p;   // bce when t=0
                float sp_neg = sp_pos - x;             // bce when t=1
                float r = __builtin_amdgcn_rcpf(1.0f + e);
                float prob = (x >= 0.0f) ? r : e * r;  // sigmoid(x)
                bool is_t = (c0 + j) == tgt;
                float om = is_t ? (1.0f - prob) : prob;        // 1 - p_t
                float af = is_t ? F_ALPHA : (1.0f - F_ALPHA);
                float bce = is_t ? sp_neg : sp_pos;
                cls_part += af * om * om * bce;
            }
        }
    }

    // ---- reduction: WMMA wave sum -> LDS -> one atomic per block ----
    float w_cls = wave_sum_wmma(cls_part);
    float w_reg = wave_sum_wmma(reg_part);
    float w_np  = wave_sum_wmma(pos ? 1.0f : 0.0f);
    int wave = tid >> 5, lane = tid & 31;
    if (lane == 0) {
        s_wcls[wave] = w_cls;
        s_wreg[wave] = w_reg;
        s_wnp[wave]  = w_np;
    }
    __syncthreads();
    if (tid == 0) {
        float bc = 0.0f, br = 0.0f, bn = 0.0f;
#pragma unroll
        for (int w = 0; w < 8; ++w) {
            bc += s_wcls[w];
            br += s_wreg[w];
            bn += s_wnp[w];
        }
        atomicAdd(&accum[0], bc);
        atomicAdd(&accum[1], br);
        atomicAdd(&accum[2], bn);
    }
}

__global__ void retina_finalize_kernel(const float* __restrict__ accum,
                                       float* __restrict__ out) {
    float cls_sum = accum[0];
    float reg_sum = accum[1];
    float np = accum[2];
    float denom = fmaxf(np, 1.0f);
    float cls_loss = (np > 0.0f) ? cls_sum / denom : cls_sum;
    float reg_loss = (np > 0.0f) ? reg_sum / denom : 0.0f;
    out[0] = cls_loss + reg_loss;
    out[1] = cls_loss;
    out[2] = reg_loss;
}

extern "C" void kernel_launch(void* const* d_in, const int* in_sizes, int n_in,
                              void* d_out, int out_size, void* d_ws,
                              size_t ws_size, hipStream_t stream) {
    const float* cls_pred  = (const float*)d_in[0];
    const float* reg_pred  = (const float*)d_in[1];
    const float* anchors   = (const float*)d_in[2];
    const float* gt_boxes  = (const float*)d_in[3];
    const int*   gt_labels = (const int*)d_in[4];
    float* out = (float*)d_out;
    float* accum = (float*)d_ws;

    retina_init_kernel<<<1, 32, 0, stream>>>(accum);
    const int grid = NB * (NA / 256);  // 3072 blocks, 256 threads (8 wave32)
    retina_loss_kernel<<<grid, 256, 0, stream>>>(cls_pred, reg_pred, anchors,
                                                 gt_boxes, gt_labels, accum);
    retina_finalize_kernel<<<1, 1, 0, stream>>>(accum, out);
}